// Encoder_76398878261733
// MI455X (gfx1250) — compile-verified
//
#include <hip/hip_runtime.h>

typedef __attribute__((ext_vector_type(2))) float v2f;
typedef __attribute__((ext_vector_type(8))) float v8f;

#define DIM 128
#define TILE_M 16
#define SW_STRIDE 136   // pad so K-row pairs land in disjoint bank ranges
#define SA_STRIDE 132   // pad so 16 row-striped b64 reads are conflict-free

// ---------------------------------------------------------------------------
// Fused 128-wide GEMM: C[M,128] = op_in(A)[M,128] @ W[128,128] + bias
//   BN_IN:    A' = relu(A*scale[col] + shift[col]) applied while staging
//   RELU_OUT: C = relu(C)
//   STATS:    accumulate per-column sum / sum-of-squares of C (pre-ReLU)
// One wave -> one 16x16 tile via V_WMMA_F32_16X16X4_F32, 8 waves cover N=128.
// ---------------------------------------------------------------------------
template <bool BN_IN, bool RELU_OUT, bool STATS>
__global__ __launch_bounds__(256) void gemm128(
    const float* __restrict__ A, const float* __restrict__ W,
    const float* __restrict__ bias, float* __restrict__ C,
    const float* __restrict__ scale, const float* __restrict__ shift,
    float* __restrict__ colsum, float* __restrict__ colsumsq, int M)
{
    __shared__ float sW[DIM * SW_STRIDE];
    __shared__ float sA[TILE_M * SA_STRIDE];

    const int tid    = threadIdx.x;
    const int lane   = tid & 31;
    const int waveId = tid >> 5;
    const int n0     = waveId * 16;
    const int mfrag  = lane & 15;          // A row within tile
    const int kb     = (lane >> 4) * 2;    // K sub-offset (ISA A/B 16x4 layout)
    const int ncol   = n0 + (lane & 15);   // output column of this lane

    // Stage W (once per block): 16384 floats across 256 threads
    for (int idx = tid; idx < DIM * DIM; idx += 256) {
        int r = idx >> 7, c = idx & 127;
        sW[r * SW_STRIDE + c] = W[idx];
    }
    const float bv = bias[ncol];
    __syncthreads();

    const int ntiles = (M + TILE_M - 1) / TILE_M;
    for (int rt = blockIdx.x; rt < ntiles; rt += gridDim.x) {
        // Stage 16x128 A tile (with optional fused BN+ReLU pre-op)
        for (int idx = tid; idx < TILE_M * DIM; idx += 256) {
            int r = idx >> 7, c = idx & 127;
            int row = rt * TILE_M + r;
            float v = (row < M) ? A[(size_t)row * DIM + c] : 0.0f;
            if (BN_IN) {
                v = fmaf(v, scale[c], shift[c]);
                v = v > 0.0f ? v : 0.0f;
            }
            sA[r * SA_STRIDE + c] = v;
        }
        __syncthreads();

        v8f acc;
#pragma unroll
        for (int r = 0; r < 8; ++r) acc[r] = bv;   // bias folded into C init

#pragma unroll
        for (int k = 0; k < DIM; k += 4) {
            v2f a, b;
            a.x = sA[mfrag * SA_STRIDE + k + kb];
            a.y = sA[mfrag * SA_STRIDE + k + kb + 1];
            b.x = sW[(k + kb) * SW_STRIDE + ncol];
            b.y = sW[(k + kb + 1) * SW_STRIDE + ncol];
            acc = __builtin_amdgcn_wmma_f32_16x16x4_f32(
                false, a, false, b, (short)0, acc, false, false);
        }

        if (STATS) {  // per-column batch stats on pre-activation y
            float s = 0.0f, s2 = 0.0f;
#pragma unroll
            for (int r = 0; r < 8; ++r) { s += acc[r]; s2 += acc[r] * acc[r]; }
            s  += __shfl_xor(s, 16, 32);   // fold rows 8..15 half-wave
            s2 += __shfl_xor(s2, 16, 32);
            if (lane < 16) {
                atomicAdd(&colsum[ncol], s);
                atomicAdd(&colsumsq[ncol], s2);
            }
        }

        if (RELU_OUT) {
#pragma unroll
            for (int r = 0; r < 8; ++r) acc[r] = acc[r] > 0.0f ? acc[r] : 0.0f;
        }

        // C layout: VGPR r, lanes<16 -> row r, lanes>=16 -> row r+8
        const int rbase = rt * TILE_M + (lane >> 4) * 8;
#pragma unroll
        for (int r = 0; r < 8; ++r) {
            int row = rbase + r;
            if (row < M) C[(size_t)row * DIM + ncol] = acc[r];
        }
        __syncthreads();
    }
}

// One wave per edge: b128 gather of a 128-float row, 4 f32 atomics per lane.
__global__ __launch_bounds__(256) void scatter_add(
    const float* __restrict__ h, const int* __restrict__ src,
    const int* __restrict__ dst, float* __restrict__ agg, int n_edges)
{
    int w = (int)((blockIdx.x * 256u + threadIdx.x) >> 5);
    int lane = threadIdx.x & 31;
    if (w >= n_edges) return;
    int s = src[w];
    int d = dst[w];
    const float4 v = *reinterpret_cast<const float4*>(h + (size_t)s * DIM + lane * 4);
    float* p = agg + (size_t)d * DIM + lane * 4;
    atomicAdd(p + 0, v.x);
    atomicAdd(p + 1, v.y);
    atomicAdd(p + 2, v.z);
    atomicAdd(p + 3, v.w);
}

__global__ void copy_vec4(const float4* __restrict__ in, float4* __restrict__ out, int n4)
{
    int i = blockIdx.x * blockDim.x + threadIdx.x;
    int stride = gridDim.x * blockDim.x;
    for (; i < n4; i += stride) out[i] = in[i];
}

__global__ void zero_f32(float* __restrict__ p, int n)
{
    int i = blockIdx.x * blockDim.x + threadIdx.x;
    if (i < n) p[i] = 0.0f;
}

__global__ void bn_finalize(const float* __restrict__ colsum,
                            const float* __restrict__ colsumsq,
                            const float* __restrict__ gamma,
                            const float* __restrict__ beta,
                            float* __restrict__ scale, float* __restrict__ shift,
                            float invM)
{
    int n = threadIdx.x;
    float mean = colsum[n] * invM;
    float var  = colsumsq[n] * invM - mean * mean;   // biased variance
    float sc   = gamma[n] * rsqrtf(var + 1e-5f);
    scale[n] = sc;
    shift[n] = beta[n] - mean * sc;
}

// One wave per node: segment-sum into [NUM_GRAPHS,128] via f32 atomics.
__global__ __launch_bounds__(256) void pool_add(
    const float* __restrict__ h, const int* __restrict__ batch,
    float* __restrict__ out, int n_nodes)
{
    int w = (int)((blockIdx.x * 256u + threadIdx.x) >> 5);
    int lane = threadIdx.x & 31;
    if (w >= n_nodes) return;
    int g = batch[w];
    const float4 v = *reinterpret_cast<const float4*>(h + (size_t)w * DIM + lane * 4);
    float* p = out + (size_t)g * DIM + lane * 4;
    atomicAdd(p + 0, v.x);
    atomicAdd(p + 1, v.y);
    atomicAdd(p + 2, v.z);
    atomicAdd(p + 3, v.w);
}

extern "C" void kernel_launch(void* const* d_in, const int* in_sizes, int n_in,
                              void* d_out, int out_size, void* d_ws, size_t ws_size,
                              hipStream_t stream)
{
    const float* x     = (const float*)d_in[0];
    const int*   edge  = (const int*)d_in[1];
    const int*   batch = (const int*)d_in[2];
    const float* Wnode = (const float*)d_in[3];
    const float* bnode = (const float*)d_in[4];
    const float* W1    = (const float*)d_in[5];
    const float* b1    = (const float*)d_in[6];
    const float* gamma = (const float*)d_in[7];
    const float* beta  = (const float*)d_in[8];
    const float* W2    = (const float*)d_in[9];
    const float* b2    = (const float*)d_in[10];
    float* out = (float*)d_out;

    const int M       = in_sizes[0] / DIM;   // 100000 nodes
    const int n_edges = in_sizes[1] / 2;     // 3.2M edges
    const size_t nf   = (size_t)M * DIM;

    float* ws       = (float*)d_ws;
    float* h        = ws;                 // [M,128]
    float* agg      = ws + nf;            // [M,128]
    float* y        = ws + 2 * nf;        // [M,128]
    float* colsum   = ws + 3 * nf;        // [128]
    float* colsumsq = colsum + DIM;       // [128]
    float* scale    = colsum + 2 * DIM;   // [128]
    float* shift    = colsum + 3 * DIM;   // [128]

    const int ntiles = (M + TILE_M - 1) / TILE_M;
    const int gemm_blocks = ntiles < 1024 ? ntiles : 1024;
    const int n4 = (int)(nf / 4);
    const int edge_blocks = (n_edges * 32 + 255) / 256;
    const int node_blocks = (M * 32 + 255) / 256;

    // h = x @ W_node + b_node
    gemm128<false, false, false><<<gemm_blocks, 256, 0, stream>>>(
        x, Wnode, bnode, h, nullptr, nullptr, nullptr, nullptr, M);

    for (int l = 0; l < 3; ++l) {
        // agg = h  (GIN: out = scatter_add + (1+eps)*h, eps = 0)
        copy_vec4<<<2048, 256, 0, stream>>>((const float4*)h, (float4*)agg, n4);
        // agg[dst] += h[src]
        scatter_add<<<edge_blocks, 256, 0, stream>>>(h, edge, edge + n_edges, agg, n_edges);
        // y = agg @ W1 + b1, with fused per-column sum / sumsq
        zero_f32<<<1, 256, 0, stream>>>(colsum, 2 * DIM);
        gemm128<false, false, true><<<gemm_blocks, 256, 0, stream>>>(
            agg, W1 + (size_t)l * DIM * DIM, b1 + l * DIM, y,
            nullptr, nullptr, colsum, colsumsq, M);
        // BN scale/shift from batch statistics
        bn_finalize<<<1, DIM, 0, stream>>>(colsum, colsumsq,
                                           gamma + l * DIM, beta + l * DIM,
                                           scale, shift, 1.0f / (float)M);
        // h = relu( relu(BN(y)) @ W2 + b2 )
        gemm128<true, true, false><<<gemm_blocks, 256, 0, stream>>>(
            y, W2 + (size_t)l * DIM * DIM, b2 + l * DIM, h,
            scale, shift, nullptr, nullptr, M);
    }

    // global_add_pool
    zero_f32<<<(out_size + 255) / 256, 256, 0, stream>>>(out, out_size);
    pool_add<<<node_blocks, 256, 0, stream>>>(h, batch, out, M);
}